// InferenceEmbeddingTable_23562190585872
// MI455X (gfx1250) — compile-verified
//
#include <hip/hip_runtime.h>
#include <hip/hip_bf16.h>

// ---------------------------------------------------------------------------
// Embedding gather + segment-sum pooling for MI455X (gfx1250).
//
// Roofline: ~420 MB of random 512B row gathers at 23.3 TB/s HBM -> ~18us
// floor; compute (105 MFLOP) is free. The segment-sum runs on
// V_WMMA_F32_16X16X4_F32: A(16x4) = one-hot "key k belongs to bag m"
// selection matrix, B(4x16) = 4 gathered rows x 16 dims, C(16x16) = 16 bags
// x 16 dims accumulator.
//
// Partition: blockIdx.x = tile of 16 consecutive segments (sorted ids ->
// contiguous key range via binary search); blockIdx.y = 32-dim slice.
// 4096 single-wave workgroups; every gathered byte read once, every output
// element written exactly once (no atomics, deterministic, no pre-zeroing).
//
// Pipeline (verified in round-3 asm: drain opens at s_wait_loadcnt 0x3e):
//   prep_load(next)   : 4 coalesced index loads issued FIRST (in-order
//                       completion => waiting on them never drains gathers)
//   issue 64 gathers  : u32 offsets -> GVS saddr form, one address VGPR
//   prep_shuffle(next): ds_bpermute broadcasts + A build + offsets, executed
//                       while the 64 gathers are in flight
//   sched_barrier(0)  : fence so the MI scheduler can't sink loads into the
//   WMMA drain        : 32 WMMAs with progressive s_wait_loadcnt
// This removes the per-block bubble where no loads were outstanding.
// ---------------------------------------------------------------------------

#define CAPACITY 2000000
#define DIM      128
#define BATCH    16384
#define NKEYS    819200

typedef __attribute__((ext_vector_type(2))) float v2f;
typedef __attribute__((ext_vector_type(8))) float v8f;

#define SCHED_FENCE() __builtin_amdgcn_sched_barrier(0)

__device__ __forceinline__ int lower_bound_i32(const int* __restrict__ s,
                                               int n, int target) {
  int lo = 0, hi = n;
  while (lo < hi) {
    int mid = (lo + hi) >> 1;
    if (s[mid] < target) lo = mid + 1; else hi = mid;
  }
  return lo;
}

// Issue the coalesced index loads for a 64-key block (2 per lane).
__device__ __forceinline__ void prep_load(const int* __restrict__ keys,
                                          const int* __restrict__ segs,
                                          int i, int lane,
                                          int& key0, int& seg0,
                                          int& key1, int& seg1) {
  __builtin_prefetch(keys + i + 512, 0, 0);
  __builtin_prefetch(segs + i + 512, 0, 0);
  key0 = keys[i + lane];
  seg0 = segs[i + lane];
  key1 = keys[i + 32 + lane];
  seg1 = segs[i + 32 + lane];
}

// Broadcast per-group rows/segments (wave32 ds_bpermute), build the one-hot
// A matrices and the u32 gather offsets for all 16 groups of a block.
__device__ __forceinline__ void prep_shuffle(int key0, int seg0,
                                             int key1, int seg1,
                                             int base, int half, int n,
                                             unsigned dcol0,
                                             v2f* __restrict__ a,
                                             unsigned* __restrict__ offA,
                                             unsigned* __restrict__ offB) {
  const unsigned rowL0 = 1u + (unsigned)(key0 % CAPACITY);
  const unsigned rowL1 = 1u + (unsigned)(key1 % CAPACITY);
  #pragma unroll
  for (int g = 0; g < 16; ++g) {
    const int sA = 4 * g + (half << 1);  // key slot 0..63 for K=2*half
    const int sB = sA + 1;               // key slot for K=2*half+1
    const unsigned rowA =
        (unsigned)__shfl((int)((g < 8) ? rowL0 : rowL1), sA & 31, 32);
    const unsigned rowB =
        (unsigned)__shfl((int)((g < 8) ? rowL0 : rowL1), sB & 31, 32);
    const int segA = __shfl((g < 8) ? seg0 : seg1, sA & 31, 32);
    const int segB = __shfl((g < 8) ? seg0 : seg1, sB & 31, 32);
    a[g].x = (segA == base + n) ? 1.0f : 0.0f;
    a[g].y = (segB == base + n) ? 1.0f : 0.0f;
    offA[g] = rowA * DIM + dcol0;        // < 2^28, fits u32
    offB[g] = rowB * DIM + dcol0;
  }
}

__global__ __launch_bounds__(32)
void emb_pool_wmma(const int* __restrict__ keys,
                   const int* __restrict__ segs,
                   const float* __restrict__ weight,
                   float* __restrict__ out) {
  const int tile = blockIdx.x;            // 16 segments per tile
  const int sgrp = blockIdx.y;            // 32 dims per slice-group
  const int base = tile * 16;
  const int lane = threadIdx.x & 31;
  const int half = lane >> 4;             // 0 -> K{0,1}, 1 -> K{2,3}
  const int n    = lane & 15;             // M (for A) / N (for B,C,D)
  const unsigned dcol0 = sgrp * 32 + n;   // first 16-dim slice (elements)

  // Contiguous key range covered by this 16-segment tile (ids are sorted).
  const int start = lower_bound_i32(segs, NKEYS, base);
  const int end   = lower_bound_i32(segs, NKEYS, base + 16);

  v8f c0 = {};  // bags x dims [sgrp*32 .. +16)
  v8f c1 = {};  // bags x dims [sgrp*32+16 .. +32)

  int i = start;
  const int nfull = (end - start) >> 6;   // number of full 64-key blocks

  if (nfull > 0) {
    v2f aC[16];
    unsigned offAC[16], offBC[16];
    v2f aN[16];
    unsigned offAN[16], offBN[16];

    // Prologue: prep block 0.
    {
      int k0, s0, k1, s1;
      prep_load(keys, segs, i, lane, k0, s0, k1, s1);
      prep_shuffle(k0, s0, k1, s1, base, half, n, dcol0, aC, offAC, offBC);
    }

    for (int blk = 0; blk < nfull; ++blk) {
      const bool has_next = (blk + 1 < nfull);

      // Next block's index loads go first: in-order return means waiting on
      // them later never forces the 64 gathers below to drain.
      int k0 = 0, s0 = 0, k1 = 0, s1 = 0;
      if (has_next) prep_load(keys, segs, i + 64, lane, k0, s0, k1, s1);

      // Issue ALL 64 gather loads for the current block.
      v2f b0[16], b1[16];
      #pragma unroll
      for (int g = 0; g < 16; ++g) {
        b0[g].x = weight[offAC[g]];
        b0[g].y = weight[offBC[g]];
        b1[g].x = weight[offAC[g] + 16];  // folds into instruction offset:64
        b1[g].y = weight[offBC[g] + 16];
      }

      // Next block's shuffles + A/offset build run while gathers are in
      // flight (ds_bpermute waits touch DScnt, not LOADcnt).
      if (has_next)
        prep_shuffle(k0, s0, k1, s1, base, half, n, dcol0, aN, offAN, offBN);

      SCHED_FENCE();

      // Drain: 32 WMMAs with progressive s_wait_loadcnt.
      #pragma unroll
      for (int g = 0; g < 16; ++g) {
        c0 = __builtin_amdgcn_wmma_f32_16x16x4_f32(false, aC[g], false, b0[g],
                                                   (short)0, c0, false, false);
        c1 = __builtin_amdgcn_wmma_f32_16x16x4_f32(false, aC[g], false, b1[g],
                                                   (short)0, c1, false, false);
      }

      SCHED_FENCE();

      // Rotate next -> current (register moves only).
      if (has_next) {
        #pragma unroll
        for (int g = 0; g < 16; ++g) {
          aC[g] = aN[g];
          offAC[g] = offAN[g];
          offBC[g] = offBN[g];
        }
      }
      i += 64;
    }
  }

  // ---- tail: 4 keys per iteration with bounds masking ----
  for (; i < end; i += 4) {
    const int kA = i + (half << 1);
    const int kB = kA + 1;

    int segA = -1, segB = -1;
    unsigned rowA = 0, rowB = 0;          // row 0 = zero sentinel (safe pad)
    if (kA < end) { segA = segs[kA]; rowA = 1u + (unsigned)(keys[kA] % CAPACITY); }
    if (kB < end) { segB = segs[kB]; rowB = 1u + (unsigned)(keys[kB] % CAPACITY); }

    v2f a;
    a.x = (segA == base + n) ? 1.0f : 0.0f;
    a.y = (segB == base + n) ? 1.0f : 0.0f;

    const unsigned offA = rowA * DIM + dcol0;
    const unsigned offB = rowB * DIM + dcol0;
    v2f b0, b1;
    b0.x = weight[offA];
    b0.y = weight[offB];
    b1.x = weight[offA + 16];
    b1.y = weight[offB + 16];

    c0 = __builtin_amdgcn_wmma_f32_16x16x4_f32(false, a, false, b0,
                                               (short)0, c0, false, false);
    c1 = __builtin_amdgcn_wmma_f32_16x16x4_f32(false, a, false, b1,
                                               (short)0, c1, false, false);
  }

  // C/D layout: VGPR r -> {M=r | M=8+r}, N = lane&15. Each output element is
  // written exactly once across the whole grid (empty bags store 0).
  #pragma unroll
  for (int r = 0; r < 8; ++r) {
    const int m = half * 8 + r;
    float* o = out + (size_t)(base + m) * DIM + sgrp * 32 + n;
    o[0]  = c0[r];
    o[16] = c1[r];
  }
}

extern "C" void kernel_launch(void* const* d_in, const int* in_sizes, int n_in,
                              void* d_out, int out_size, void* d_ws, size_t ws_size,
                              hipStream_t stream) {
  const int*   keys   = (const int*)d_in[0];
  const int*   segs   = (const int*)d_in[1];
  const float* weight = (const float*)d_in[2];
  float*       out    = (float*)d_out;

  dim3 grid(BATCH / 16, DIM / 32, 1);   // 1024 tiles x 4 dim-slices
  dim3 block(32, 1, 1);                 // one wave32 per workgroup
  hipLaunchKernelGGL(emb_pool_wmma, grid, block, 0, stream,
                     keys, segs, weight, out);
}